// MultiBoxHeads_46411416600799
// MI455X (gfx1250) — compile-verified
//
#include <hip/hip_runtime.h>

#define PRI    3000   // number of priors
#define NCLS   201
#define C1     200    // foreground classes
#define NB     16     // batch
#define TOPN   100
#define SM_TILE 32    // priors per softmax block (2 waves x 16 WMMA rows)
#define SM_BLK  64

typedef __attribute__((ext_vector_type(4))) unsigned int u32x4;
typedef __attribute__((ext_vector_type(8))) int          i32x8;
typedef __attribute__((ext_vector_type(4))) int          i32x4;
typedef __attribute__((ext_vector_type(2))) float        v2f;
typedef __attribute__((ext_vector_type(8))) float        v8f;

// ---- wave32 (value,key) argmax: higher value wins, ties -> lower key (jnp.argmax order)
__device__ __forceinline__ void waveArgMax(float &v, int &k) {
#pragma unroll
  for (int off = 16; off > 0; off >>= 1) {
    float ov = __shfl_xor(v, off, 32);
    int   ok = __shfl_xor(k, off, 32);
    if (ov > v || (ov == v && ok < k)) { v = ov; k = ok; }
  }
}

// ---- block-wide argmax for exactly 1024 threads (32 waves); broadcasts result to all
__device__ __forceinline__ void blockArgMax1024(float &v, int &k, float *sv, int *sk) {
  waveArgMax(v, k);
  const int lane = threadIdx.x & 31, wid = threadIdx.x >> 5;
  if (lane == 0) { sv[wid] = v; sk[wid] = k; }
  __syncthreads();
  if (wid == 0) {
    float vv = sv[lane]; int kk = sk[lane];
    waveArgMax(vv, kk);
    if (lane == 0) { sv[0] = vv; sk[0] = kk; }
  }
  __syncthreads();
  v = sv[0]; k = sk[0];
  __syncthreads();   // sv/sk safe for reuse
}

// ============ kernel 0: decode prior boxes ============
__global__ __launch_bounds__(256) void decode_kernel(const float *__restrict__ deltas,
                                                     const float *__restrict__ priors,
                                                     float4 *__restrict__ dec) {
  const int idx = blockIdx.x * 256 + threadIdx.x;
  if (idx >= NB * PRI) return;
  const int p = idx % PRI;
  const float4 d  = ((const float4 *)deltas)[idx];
  const float4 pr = ((const float4 *)priors)[p];
  const float cx = d.x * 0.1f * pr.z + pr.x;
  const float cy = d.y * 0.1f * pr.w + pr.y;
  const float w  = expf(d.z * 0.2f) * pr.z;
  const float h  = expf(d.w * 0.2f) * pr.w;
  dec[idx] = make_float4(cx - 0.5f * w, cy - 0.5f * h, cx + 0.5f * w, cy + 0.5f * h);
}

// ============ kernel 1: softmax + threshold + transpose to [b][c][p] ============
// Row sums of exp() run on the matrix pipe: V_WMMA_F32_16X16X4_F32 with an all-ones B
// gives D[m][n] = sum_k A[m][k] in full fp32. Pad rows are zero-filled in LDS (WMMA row m
// of D depends only on row m of A, so pads can't pollute valid rows) and the odd 201st
// column is a peeled, branch-free tail -> the 50-iteration hot loop has no predication.
__global__ __launch_bounds__(SM_BLK) void softmax_tr_kernel(const float *__restrict__ obj,
                                                            float *__restrict__ scores) {
  __shared__ float tile[SM_TILE * NCLS];   // 32*201*4 = 25728 B
  __shared__ float rmax[SM_TILE];
  __shared__ float rinv[SM_TILE];
  const int tilesPerImg = (PRI + SM_TILE - 1) / SM_TILE;   // 94
  const int b      = blockIdx.x / tilesPerImg;
  const int p0     = (blockIdx.x % tilesPerImg) * SM_TILE;
  const int nvalid = min(SM_TILE, PRI - p0);               // 32, or 24 for the last tile
  const float *src = obj + ((size_t)b * PRI + p0) * NCLS;
  for (int i = threadIdx.x; i < nvalid * NCLS; i += SM_BLK) tile[i] = src[i];  // coalesced
  for (int i = nvalid * NCLS + threadIdx.x; i < SM_TILE * NCLS; i += SM_BLK) tile[i] = 0.f;
  __syncthreads();

  const int lane = threadIdx.x & 31;
  const int wid  = threadIdx.x >> 5;            // wave 0 -> rows 0..15, wave 1 -> rows 16..31
  const int row  = (wid << 4) + (lane & 15);    // tile-local row for this lane (A-matrix M)
  const int kh   = lane >> 4;                   // A-layout: lanes 0-15 hold K0,K1; 16-31 hold K2,K3
  const float *trow = &tile[row * NCLS];

  // per-row max: lane pair (L, L+16) scans even/odd columns, combine with xor-16 shuffle
  float m = -3.4e38f;
  for (int c = kh; c < NCLS; c += 2) m = fmaxf(m, trow[c]);
  m = fmaxf(m, __shfl_xor(m, 16, 32));

#if __has_builtin(__builtin_amdgcn_wmma_f32_16x16x4_f32)
  v8f acc = {};
  v2f bones; bones.x = 1.f; bones.y = 1.f;      // B = ones(4x16): layout-independent
  for (int kk = 0; kk < NCLS - 1; kk += 4) {    // 50 branch-free iterations, K = 0..199
    const int c0 = kk + 2 * kh;
    v2f a;
    a.x = expf(trow[c0]     - m);
    a.y = expf(trow[c0 + 1] - m);
    acc = __builtin_amdgcn_wmma_f32_16x16x4_f32(false, a, false, bones,
                                                (short)0, acc, false, false);
  }
  {                                             // peeled tail: lone column K = 200
    const float e = expf(trow[NCLS - 1] - m);   // unconditional expf, then a cndmask
    v2f a; a.x = (kh == 0) ? e : 0.f; a.y = 0.f;
    acc = __builtin_amdgcn_wmma_f32_16x16x4_f32(false, a, false, bones,
                                                (short)0, acc, false, false);
  }
  // D layout: VGPR r -> M=r (lanes 0-15) / M=8+r (lanes 16-31); any column is the row sum
  if (lane == 0 || lane == 16) {
    const int rbase = (wid << 4) + ((lane >> 4) << 3);
#pragma unroll
    for (int r = 0; r < 8; ++r)
      if (rbase + r < nvalid) rinv[rbase + r] = 1.f / acc[r];
  }
#else
  float ssum = 0.f;
  for (int c = kh; c < NCLS; c += 2) ssum += expf(trow[c] - m);
  ssum += __shfl_xor(ssum, 16, 32);
  if (lane < 16 && row < nvalid) rinv[row] = 1.f / ssum;
#endif
  if (lane < 16 && row < nvalid) rmax[row] = m;
  __syncthreads();

  float *dst = scores + (size_t)b * C1 * PRI + p0;
  for (int i = threadIdx.x; i < C1 * nvalid; i += SM_BLK) {  // coalesced transposed store
    const int c = i / nvalid, pp = i % nvalid;
    const float pr = expf(tile[pp * NCLS + (c + 1)] - rmax[pp]) * rinv[pp];
    dst[(size_t)c * PRI + pp] = (pr > 0.02f) ? pr : -1.f;
  }
}

// ============ kernel 2: initial per-class (max, argmax) — one wave per (b,c) ============
__global__ __launch_bounds__(256) void classmax_kernel(const float *__restrict__ scores,
                                                       float *__restrict__ cmax,
                                                       int *__restrict__ carg) {
  const int gw   = (blockIdx.x * 256 + threadIdx.x) >> 5;
  const int lane = threadIdx.x & 31;
  if (gw >= NB * C1) return;
  const float *s = scores + (size_t)gw * PRI;
  float v = -3.4e38f; int k = 0x7FFFFFFF;
  for (int p = lane; p < PRI; p += 32) {
    const float x = s[p];
    if (x > v) { v = x; k = p; }     // strictly-greater keeps lowest p on ties
  }
  waveArgMax(v, k);
  if (lane == 0) { cmax[gw] = v; carg[gw] = k; }
}

// ============ kernel 3: sequential NMS, one workgroup per image ============
__global__ __launch_bounds__(1024) void nms_kernel(const float4 *__restrict__ dec,
                                                   float *__restrict__ scores,
                                                   const float *__restrict__ cmaxg,
                                                   const int *__restrict__ cargg,
                                                   float *__restrict__ out) {
  __shared__ float4 sbox[PRI];     // 48 KB decoded boxes (320 KB WGP LDS -> fits easily)
  __shared__ float  scmax[C1];
  __shared__ int    scarg[C1];
  __shared__ float  sv[32];
  __shared__ int    sk[32];

  const int b   = blockIdx.x;
  const int tid = threadIdx.x;

#if __has_builtin(__builtin_amdgcn_tensor_load_to_lds)
  {
    // Tensor Data Mover: DMA the 3000x4-float box table (48000 B) global -> LDS.
    const unsigned long long ga = (unsigned long long)(const void *)(dec + (size_t)b * PRI);
    const unsigned ldsOff = (unsigned)(unsigned long long)(void *)&sbox[0];
    u32x4 g0;
    g0[0] = 1u;                                                  // count=1, user descriptor
    g0[1] = ldsOff;                                              // lds_addr
    g0[2] = (unsigned)(ga & 0xFFFFFFFFull);                      // global_addr[31:0]
    g0[3] = (unsigned)((ga >> 32) & 0x1FFFFFFull) | (2u << 30);  // global_addr[56:32], type=2
    i32x8 g1;
    g1[0] = 0x00020000;                                          // data_size=2 -> 4-byte elems
    g1[1] = (int)(((unsigned)(PRI * 4) & 0xFFFFu) << 16);        // tensor_dim0[15:0]
    g1[2] = (int)(((unsigned)(PRI * 4) >> 16) | (1u << 16));     // tensor_dim0[31:16] | tensor_dim1=1
    g1[3] = (int)((unsigned)(PRI * 4) << 16);                    // tile_dim0 = 12000
    g1[4] = 1;                                                   // tile_dim1 = 1, tile_dim2 = 0
    g1[5] = PRI * 4;                                             // tensor_dim0_stride = 12000
    g1[6] = 0;
    g1[7] = 0;
    i32x4 zz4; zz4[0] = 0; zz4[1] = 0; zz4[2] = 0; zz4[3] = 0;   // groups 2/3 unused
    i32x8 zz8;
#pragma unroll
    for (int i = 0; i < 8; ++i) zz8[i] = 0;
    if (__builtin_amdgcn_readfirstlane((int)tid) == 0) {         // scalar branch: wave 0 only
      __builtin_amdgcn_tensor_load_to_lds(g0, g1, zz4, zz4, zz8, 0);
    }
  }
  __builtin_amdgcn_s_wait_tensorcnt(0);
#else
  for (int p = tid; p < PRI; p += 1024) sbox[p] = dec[(size_t)b * PRI + p];
#endif
  if (tid < C1) { scmax[tid] = cmaxg[b * C1 + tid]; scarg[tid] = cargg[b * C1 + tid]; }
  __syncthreads();

  float *simg = scores + (size_t)b * C1 * PRI;
  float *o    = out + b * (TOPN * 6);

  for (int t = 0; t < TOPN; ++t) {
    // ---- global pick over per-class maxima; key = p*200 + c matches jnp.argmax flat order
    float v = -3.4e38f; int key = 0x7FFFFFFF;
    if (tid < C1) { v = scmax[tid]; key = scarg[tid] * C1 + tid; }
    blockArgMax1024(v, key, sv, sk);
    const int csel = key % C1;
    const int psel = key / C1;
    const float4 bs = sbox[psel];

    if (tid == 0) {
      const bool ok = v > 0.02f;
      o[t * 6 + 0] = ok ? fminf(fmaxf(bs.x, 0.f), 1.f) : 0.f;
      o[t * 6 + 1] = ok ? fminf(fmaxf(bs.y, 0.f), 1.f) : 0.f;
      o[t * 6 + 2] = ok ? fminf(fmaxf(bs.z, 0.f), 1.f) : 0.f;
      o[t * 6 + 3] = ok ? fminf(fmaxf(bs.w, 0.f), 1.f) : 0.f;
      o[t * 6 + 4] = ok ? v : 0.f;
      o[t * 6 + 5] = ok ? (float)(csel + 1) : 0.f;
    }

    // ---- suppress within selected class (cross-class IoU is exactly 0 by construction)
    //      and rebuild that class's (max, argmax) in the same pass.
    const float areaS = (bs.z - bs.x) * (bs.w - bs.y);
    float *sc = simg + (size_t)csel * PRI;
    float nv = -3.4e38f; int nk = 0x7FFFFFFF;
    for (int p = tid; p < PRI; p += 1024) {      // fixed per-thread ownership of sc[p]
      float x = sc[p];
      const float4 bp = sbox[p];
      const float iw = fminf(bp.z, bs.z) - fmaxf(bp.x, bs.x);
      const float ih = fminf(bp.w, bs.w) - fmaxf(bp.y, bs.y);
      const float inter = fmaxf(iw, 0.f) * fmaxf(ih, 0.f);
      const float areaP = (bp.z - bp.x) * (bp.w - bp.y);
      const float iou = inter / (areaP + areaS - inter);
      if (iou > 0.45f || p == psel) { x = -1.f; sc[p] = x; }
      if (x > nv) { nv = x; nk = p; }
    }
    blockArgMax1024(nv, nk, sv, sk);
    if (tid == 0) { scmax[csel] = nv; scarg[csel] = nk; }
    __syncthreads();
  }
}

extern "C" void kernel_launch(void *const *d_in, const int *in_sizes, int n_in,
                              void *d_out, int out_size, void *d_ws, size_t ws_size,
                              hipStream_t stream) {
  const float *deltas = (const float *)d_in[0];  // (16, 3000, 4)
  const float *obj    = (const float *)d_in[1];  // (16, 3000, 201)
  const float *priors = (const float *)d_in[2];  // (3000, 4)

  // workspace layout (~39.2 MB): dec | scores | cmax | carg
  float *dec    = (float *)d_ws;                       // 16*3000*4 floats
  float *scores = dec + (size_t)NB * PRI * 4;          // 16*200*3000 floats, [b][c][p]
  float *cmax   = scores + (size_t)NB * C1 * PRI;      // 3200 floats
  int   *carg   = (int *)(cmax + NB * C1);             // 3200 ints

  const int tilesPerImg = (PRI + SM_TILE - 1) / SM_TILE;   // 94
  decode_kernel<<<(NB * PRI + 255) / 256, 256, 0, stream>>>(deltas, priors, (float4 *)dec);
  softmax_tr_kernel<<<NB * tilesPerImg, SM_BLK, 0, stream>>>(obj, scores);
  classmax_kernel<<<(NB * C1) / 8, 256, 0, stream>>>(scores, cmax, carg);
  nms_kernel<<<NB, 1024, 0, stream>>>((const float4 *)dec, scores, cmax, carg, (float *)d_out);
}